// SCL_Dec_24077586662017
// MI455X (gfx1250) — compile-verified
//
#include <hip/hip_runtime.h>

// SCL polar decoder, N=256, LIST=8 (16 path slots), batch=256.
// One workgroup (256 threads = 8 wave32) per codeword; all state LDS-resident:
//   - compact per-stage LLR buffers: llr[16 paths][255 (+pad)] f32   (~16 KB)
//   - bit-packed partial sums:       ub[16 paths][9 stages * 8 u32]  (~4.6 KB)
// Path sort/clone at each info bit = one-hot 16x16 permutation applied to the
// 16x256 f32 LLR state via chained V_WMMA_F32_16X16X4_F32 (exact in f32).
// Two independent accumulator chains are interleaved to hide the WMMA->WMMA
// accumulator RAW hazard; all B fragments are fetched before the chains start.

#define N_        256
#define NPATH     16
#define LLR_MAXC  30.0f
#define LLR_STR   257          // row pad to break LDS bank alignment
#define UB_WORDS  72           // 9 stages * 8 words (256 bits)
#define UB_TOTAL  (NPATH * UB_WORDS)   // 1152

#if __has_builtin(__builtin_amdgcn_wmma_f32_16x16x4_f32)
#define HAVE_WMMA_F32X4 1
#else
#define HAVE_WMMA_F32X4 0
#endif

typedef __attribute__((ext_vector_type(2))) float v2f;
typedef __attribute__((ext_vector_type(8))) float v8f;

__device__ __forceinline__ float clip30(float x) {
  return fminf(fmaxf(x, -LLR_MAXC), LLR_MAXC);
}
__device__ __forceinline__ float logaddexp_(float a, float b) {
  return fmaxf(a, b) + log1pf(__expf(-fabsf(a - b)));
}
__device__ __forceinline__ float softplus_(float x) {
  return fmaxf(x, 0.0f) + log1pf(__expf(-fabsf(x)));
}
__device__ __forceinline__ float cn_(float x, float y) {   // boxplus
  float xc = clip30(x), yc = clip30(y);
  return logaddexp_(xc + yc, 0.0f) - logaddexp_(xc, yc);
}

__global__ __launch_bounds__(256) void scl_decode_kernel(
    const float* __restrict__ llr_ch, float* __restrict__ out) {
  __shared__ float    ch[N_];                 // -llr_ch (stage-8, path-invariant)
  __shared__ float    llr[NPATH][LLR_STR];    // compact stage buffers, off(s)=2^s-1
  __shared__ unsigned ub[NPATH][UB_WORDS];    // uhat bits: [path][stage*8 + word]
  __shared__ float    pm[NPATH];
  __shared__ int      rank_l[NPATH];
  __shared__ int      order_l[NPATH];
  __shared__ int      best_l;

  const int tid = threadIdx.x;
  const int b   = blockIdx.x;

  // ---- init ----
  ch[tid] = -llr_ch[b * N_ + tid];
  for (int idx = tid; idx < UB_TOTAL; idx += 256)
    ub[idx / UB_WORDS][idx % UB_WORDS] = 0u;
  if (tid < NPATH) pm[tid] = (tid == 0 || tid == 8) ? 0.0f : LLR_MAXC * (float)N_;
  __syncthreads();

  // f: child(s-1) <- boxplus of parent(s) halves
  auto f_op = [&](int s) {
    const int half = 1 << (s - 1);
    const int offc = half - 1, offp = 2 * half - 1;
    for (int idx = tid; idx < NPATH * half; idx += 256) {
      const int p = idx >> (s - 1), i = idx & (half - 1);
      float a, bb;
      if (s == 8) { a = ch[i]; bb = ch[i + 128]; }
      else        { a = llr[p][offp + i]; bb = llr[p][offp + half + i]; }
      llr[p][offc + i] = cn_(a, bb);
    }
  };
  // g: child(s-1) <- (1-2*u)*left + right, u from bit-packed uhat at stage s-1
  auto g_op = [&](int s, int base) {
    const int half = 1 << (s - 1);
    const int offc = half - 1, offp = 2 * half - 1;
    for (int idx = tid; idx < NPATH * half; idx += 256) {
      const int p = idx >> (s - 1), i = idx & (half - 1);
      float a, bb;
      if (s == 8) { a = ch[i]; bb = ch[i + 128]; }
      else        { a = llr[p][offp + i]; bb = llr[p][offp + half + i]; }
      const int pos = base + i;
      const unsigned w = ub[p][(s - 1) * 8 + (pos >> 5)];
      const float ul = (float)((w >> (pos & 31)) & 1u);
      llr[p][offc + i] = (1.0f - 2.0f * ul) * a + bb;
    }
  };
  // combine: stage-s bits [base,base+2^s) <- (ul^ur, ur) from stage s-1
  auto combine_op = [&](int s, int base) {
    const int half = 1 << (s - 1);
    if (half >= 32) {                          // word-aligned, word-parallel
      const int hw = half >> 5;
      for (int idx = tid; idx < NPATH * hw; idx += 256) {
        const int p = idx / hw, w = idx % hw, wl = (base >> 5) + w;
        const unsigned L = ub[p][(s - 1) * 8 + wl];
        const unsigned R = ub[p][(s - 1) * 8 + wl + hw];
        ub[p][s * 8 + wl]      = L ^ R;
        ub[p][s * 8 + wl + hw] = R;
      }
    } else if (tid < NPATH) {                  // sub-word, one thread per path
      const int p = tid, size = half << 1;
      const int wi = base >> 5, bo = base & 31;
      const unsigned hm = (1u << half) - 1u;
      const unsigned sm = (size == 32) ? 0xFFFFFFFFu : ((1u << size) - 1u);
      const unsigned prev = ub[p][(s - 1) * 8 + wi];
      const unsigned bits = (prev >> bo) & sm;
      const unsigned ul = bits & hm, ur = (bits >> half) & hm;
      const unsigned nb = (ul ^ ur) | (ur << half);
      const unsigned old = ub[p][s * 8 + wi];
      ub[p][s * 8 + wi] = (old & ~(sm << bo)) | (nb << bo);
    }
  };

#pragma unroll 1
  for (int lo = 0; lo < N_; ++lo) {
    // ---- LLR propagation down to stage 0 ----
    if (lo == 0) {
#pragma unroll 1
      for (int s = 8; s >= 1; --s) { f_op(s); __syncthreads(); }
    } else {
      const int t = __ffs(lo) - 1;             // ctz
      g_op(t + 1, lo - (1 << t)); __syncthreads();
#pragma unroll 1
      for (int s = t; s >= 1; --s) { f_op(s); __syncthreads(); }
    }

    // ---- leaf ----
    if (lo < 128) {                            // frozen: penalty only
      if (tid < NPATH) pm[tid] += softplus_(-clip30(llr[tid][0]));
      __syncthreads();
    } else {                                   // info bit: fork + sort + clone
      if (tid < NPATH) {
        const float lc = clip30(llr[tid][0]);
        pm[tid] += (tid < 8) ? softplus_(-lc) : softplus_(lc);
        if (tid >= 8) ub[tid][lo >> 5] |= (1u << (lo & 31));   // stage 0, u=1
      }
      __syncthreads();
      if (tid < NPATH) {                       // stable argsort via rank
        const float mq = pm[tid];
        int r = 0;
        for (int q = 0; q < NPATH; ++q) {
          const float mr = pm[q];
          r += (mr < mq) || (mr == mq && q < tid);
        }
        rank_l[tid] = r;
        order_l[r] = tid;
      }
      __syncthreads();

      // -------- read phase: gather everything into registers --------
      float pmv = 0.0f;
      if (tid < NPATH) pmv = pm[order_l[tid & 7]];
      // uhat gather: static per-k register indices (no dynamic reg addressing)
      unsigned ubv[5];
#pragma unroll
      for (int k = 0; k < 5; ++k) {
        const int idx = tid + k * 256;
        if (idx < UB_TOTAL)
          ubv[k] = ub[order_l[(idx / UB_WORDS) & 7]][idx % UB_WORDS];
      }

#if HAVE_WMMA_F32X4
      // D = P x S : P[m][q] = (rank[q] == (m&7)) encodes sort+clone of 8 best.
      const int lane = tid & 31, wave = tid >> 5;
      const int m15 = lane & 15, hi = lane >> 4;
      const int col0 = (2 * wave) * 16 + m15;
      const int col1 = (2 * wave + 1) * 16 + m15;
      // Materialize all fragments first: A (shared by both tiles) + both B sets.
      v2f af[4], b0[4], b1[4];
#pragma unroll
      for (int c = 0; c < 4; ++c) {            // ISA 16x4 f32 A/B lane layout
        const int kx = 4 * c + 2 * hi;
        af[c].x = (rank_l[kx]     == (m15 & 7)) ? 1.0f : 0.0f;
        af[c].y = (rank_l[kx + 1] == (m15 & 7)) ? 1.0f : 0.0f;
        b0[c].x = llr[kx][col0];
        b0[c].y = llr[kx + 1][col0];
        b1[c].x = llr[kx][col1];
        b1[c].y = llr[kx + 1][col1];
      }
      // Two independent accumulator chains, interleaved to hide WMMA RAW hazard.
      v8f acc0 = {0.f, 0.f, 0.f, 0.f, 0.f, 0.f, 0.f, 0.f};
      v8f acc1 = {0.f, 0.f, 0.f, 0.f, 0.f, 0.f, 0.f, 0.f};
#pragma unroll
      for (int c = 0; c < 4; ++c) {
        acc0 = __builtin_amdgcn_wmma_f32_16x16x4_f32(
            false, af[c], false, b0[c], (short)0, acc0, false, false);
        acc1 = __builtin_amdgcn_wmma_f32_16x16x4_f32(
            false, af[c], false, b1[c], (short)0, acc1, false, false);
      }
#else
      float lv[16];
#pragma unroll
      for (int k = 0; k < 16; ++k) {
        const int idx = tid + k * 256;
        lv[k] = llr[order_l[(idx >> 8) & 7]][idx & 255];
      }
#endif
      __syncthreads();

      // -------- write phase --------
      if (tid < NPATH) pm[tid] = pmv;
#pragma unroll
      for (int k = 0; k < 5; ++k) {
        const int idx = tid + k * 256;
        if (idx < UB_TOTAL)
          ub[idx / UB_WORDS][idx % UB_WORDS] = ubv[k];
      }
#if HAVE_WMMA_F32X4
#pragma unroll
      for (int j = 0; j < 8; ++j) {
        llr[j + 8 * hi][col0] = acc0[j];
        llr[j + 8 * hi][col1] = acc1[j];
      }
#else
#pragma unroll
      for (int k = 0; k < 16; ++k) {
        const int idx = tid + k * 256;
        llr[idx >> 8][idx & 255] = lv[k];
      }
#endif
      __syncthreads();
    }

    // ---- partial-sum combines (one per trailing 1-bit of lo) ----
    const int ones = __ffs(~lo) - 1;
#pragma unroll 1
    for (int c = 1; c <= ones; ++c) {
      combine_op(c, lo - ((1 << c) - 1));
      __syncthreads();
    }
  }

  // ---- emit info bits of best path ----
  if (tid == 0) {
    int best = 0; float bm = pm[0];
    for (int p = 1; p < NPATH; ++p)
      if (pm[p] < bm) { bm = pm[p]; best = p; }
    best_l = best;
  }
  __syncthreads();
  if (tid < 128) {
    const int pos = 128 + tid;
    const unsigned w = ub[best_l][pos >> 5];   // stage 0 words
    out[b * 128 + tid] = (float)((w >> (pos & 31)) & 1u);
  }
}

extern "C" void kernel_launch(void* const* d_in, const int* in_sizes, int n_in,
                              void* d_out, int out_size, void* d_ws, size_t ws_size,
                              hipStream_t stream) {
  const float* llr_ch = (const float*)d_in[0];
  float* out = (float*)d_out;
  const int batch = in_sizes[0] / N_;   // 256
  scl_decode_kernel<<<batch, 256, 0, stream>>>(llr_ch, out);
}